// Ising_sampling_89928025243872
// MI455X (gfx1250) — compile-verified
//
#include <hip/hip_runtime.h>

typedef float v2f __attribute__((ext_vector_type(2)));
typedef float v8f __attribute__((ext_vector_type(8)));

// tanh via native exp2/rcp: tanh(z) = 1 - 2/(exp(2z)+1); saturates correctly at +-inf.
__device__ __forceinline__ float fast_tanh(float z) {
    float e = __builtin_amdgcn_exp2f(z * 2.8853900817779268f); // exp(2z)
    return 1.0f - 2.0f * __builtin_amdgcn_rcpf(e + 1.0f);
}

// Exchange halves of the wave32: lane l <-> lane l^16. ds_swizzle SWAPX16
// (group-of-32: xor=0x10, and=0x1f -> offset 0x401F). No LDS memory touched.
__device__ __forceinline__ float half_swap(float v) {
    return __int_as_float(__builtin_amdgcn_ds_swizzle(__float_as_int(v), 0x401F));
}

// M[p][q] from strict-lower-triangle Q, symmetrized (M = S^T + S, zero diagonal).
__device__ __forceinline__ float Mval(const float* __restrict__ Q, int p, int q) {
    if (p == q) return 0.0f;
    int i = p > q ? p : q;
    int j = p > q ? q : p;
    return Q[i * (i - 1) / 2 + j];
}

// g^T = M * x^T using 4 chained V_WMMA_F32_16X16X4_F32 (K=16 total).
// E-storage: slot (r, half) holds dim m = r + 8*half, batch = lane%16 (both halves).
// B-operand slot (v, half) needs dim k = 4*(v>>1) + 2*half + (v&1).
// Cross-half slots come from 4 combined swaps: swap(hi?x[a]:x[b]) yields
// partner-x[a] on half0 lanes AND partner-x[b] on half1 lanes in one DS op.
__device__ __forceinline__ v8f matmul16(const v2f* __restrict__ av,
                                        const float* __restrict__ x, bool hi) {
    float s02 = half_swap(hi ? x[0] : x[2]);
    float s13 = half_swap(hi ? x[1] : x[3]);
    float s46 = half_swap(hi ? x[4] : x[6]);
    float s57 = half_swap(hi ? x[5] : x[7]);

    v2f b0 = { hi ? s02 : x[0], hi ? s13 : x[1] };  // K slice 0..3
    v2f b1 = { hi ? s46 : x[4], hi ? s57 : x[5] };  // K slice 4..7
    v2f b2 = { hi ? x[2] : s02, hi ? x[3] : s13 };  // K slice 8..11
    v2f b3 = { hi ? x[6] : s46, hi ? x[7] : s57 };  // K slice 12..15

    v8f c = {};
    c = __builtin_amdgcn_wmma_f32_16x16x4_f32(false, av[0], false, b0, (short)0, c, false, false);
    c = __builtin_amdgcn_wmma_f32_16x16x4_f32(false, av[1], false, b1, (short)0, c, false, false);
    c = __builtin_amdgcn_wmma_f32_16x16x4_f32(false, av[2], false, b2, (short)0, c, false, false);
    c = __builtin_amdgcn_wmma_f32_16x16x4_f32(false, av[3], false, b3, (short)0, c, false, false);
    return c;
}

__global__ __launch_bounds__(256) void ising_leapfrog_kernel(
    const float* __restrict__ Q, const float* __restrict__ delta,
    const float* __restrict__ y0, const float* __restrict__ vn,
    float* __restrict__ xout, float* __restrict__ eout, int batch) {
    const int lane = threadIdx.x & 31;
    const int bl   = lane & 15;
    const bool hi  = (lane & 16) != 0;
    const int wave = blockIdx.x * (blockDim.x >> 5) + (threadIdx.x >> 5);
    const int tile = wave << 4;
    if (tile >= batch) return;  // wave-uniform; grid exactly covers batch anyway
    const int row = tile + bl;
    const int mofs = hi ? 8 : 0;

    // A operand: M in A-layout (row m = lane%16; K across VGPR pair + half). Constant.
    v2f av[4];
#pragma unroll
    for (int j = 0; j < 4; ++j) {
        int k0 = 4 * j + (hi ? 2 : 0);
        av[j].x = Mval(Q, bl, k0);
        av[j].y = Mval(Q, bl, k0 + 1);
    }
    float dl[8];
#pragma unroll
    for (int r = 0; r < 8; ++r) dl[r] = delta[r + mofs];

    float y[8], x[8], vel[8], f[8];
#pragma unroll
    for (int r = 0; r < 8; ++r) {
        y[r] = y0[(size_t)row * 16 + r + mofs];
        x[r] = fast_tanh(y[r]);
    }
    {
        v8f g = matmul16(av, x, hi);
#pragma unroll
        for (int r = 0; r < 8; ++r) f[r] = (dl[r] + g[r]) * (1.0f - x[r] * x[r]);
    }

    for (int layer = 0; layer < 3; ++layer) {
        const float* v0 = vn + (size_t)layer * batch * 16;
#pragma unroll
        for (int r = 0; r < 8; ++r) vel[r] = v0[(size_t)row * 16 + r + mofs];
#pragma unroll 2
        for (int step = 0; step < 10; ++step) {
            float vh[8];
#pragma unroll
            for (int r = 0; r < 8; ++r) {
                vh[r] = fmaf(0.5f, f[r], vel[r]);  // vel_half = vel + 0.5*force(x)
                y[r] += vh[r];                      // y += vel_half
                x[r] = fast_tanh(y[r]);             // x = tanh(y)
            }
            v8f g = matmul16(av, x, hi);            // force(x_new), reused next step
#pragma unroll
            for (int r = 0; r < 8; ++r) {
                f[r]   = (dl[r] + g[r]) * (1.0f - x[r] * x[r]);
                vel[r] = fmaf(0.5f, f[r], vh[r]);   // vel = vel_half + 0.5*force(x)
            }
        }
    }

    // Straight-through output: forward value = (hard - tanh(y)) + tanh(y), hard = y>0 ? 1 : -1
    float xo[8];
#pragma unroll
    for (int r = 0; r < 8; ++r) {
        float hard = y[r] > 0.0f ? 1.0f : -1.0f;
        xo[r] = (hard - x[r]) + x[r];
        xout[(size_t)row * 16 + r + mofs] = xo[r];
    }

    // energy = sum_m (0.5*(x_out@M)[m] + delta[m]) * x_out[m]
    v8f h = matmul16(av, xo, hi);
    float s = 0.0f;
#pragma unroll
    for (int r = 0; r < 8; ++r) s += fmaf(0.5f, h[r], dl[r]) * xo[r];
    s += half_swap(s);  // combine dim slots 0..7 (half0) with 8..15 (half1)
    if (!hi) eout[tile + lane] = s;
}

extern "C" void kernel_launch(void* const* d_in, const int* in_sizes, int n_in,
                              void* d_out, int out_size, void* d_ws, size_t ws_size,
                              hipStream_t stream) {
    // inputs order: inputs(unused), Q, delta, y0, vel_noise
    const float* Q     = (const float*)d_in[1];
    const float* delta = (const float*)d_in[2];
    const float* y0    = (const float*)d_in[3];
    const float* vn    = (const float*)d_in[4];
    int batch = in_sizes[3] / 16;

    float* xout = (float*)d_out;
    float* eout = xout + (size_t)batch * 16;

    int waves  = (batch + 15) / 16;          // 16 batch rows per wave32
    int blocks = (waves + 7) / 8;            // 8 waves (256 threads) per block
    ising_leapfrog_kernel<<<blocks, 256, 0, stream>>>(Q, delta, y0, vn, xout, eout, batch);
}